// YoloLossNew_90709709291828
// MI455X (gfx1250) — compile-verified
//
#include <hip/hip_runtime.h>
#include <stdint.h>

// ---- YOLO constants (TRAIN_IMAGE_SIZE=448, GRID_NUM=7) ----
#define GRID_LEN_F 64.0f      // 448/7
#define HALF_IMG_F 224.0f     // 0.5*448
#define L_COORD_F  5.0f
#define L_NOOBJ_F  0.5f
#define NVEC       26

#define CHUNK     128         // cells per TDM tile (128*26*4 = 13312 B per tensor)
#define P1_BLOCK  128         // 4 waves (wave32)
#define P2_BLOCK  256

typedef unsigned int v4u __attribute__((ext_vector_type(4)));
typedef int          v8i __attribute__((ext_vector_type(8)));
typedef int          v4i __attribute__((ext_vector_type(4)));

// --------------------------------------------------------------------------
// Tensor Data Mover: 1-D (single-row 2-D tile) load of `nelem` fp32 elements
// from global to LDS. Descriptor bitfields per CDNA5 ISA
// sections 8.3/8.4. amdgpu-toolchain 6-arg builtin form:
//   (uint32x4 g0, int32x8 g1, int32x4 g2, int32x4 g3, int32x8 g4, i32 cpol)
// --------------------------------------------------------------------------
__device__ __forceinline__ void tdm_load_1d(uint32_t lds_off, const void* gptr,
                                            uint32_t nelem)
{
    uint64_t ga = (uint64_t)(uintptr_t)gptr;
    v4u g0;
    g0.x = 1u;                                   // count=1 (valid user D#)
    g0.y = lds_off;                              // lds_addr (bytes)
    g0.z = (uint32_t)ga;                         // global_addr[31:0]
    g0.w = (uint32_t)(ga >> 32) | (2u << 30);    // global_addr[56:32] | type=2

    v8i g1;
    g1[0] = (int)(2u << 16);                     // wg_mask=0, data_size=2 (4 B)
    g1[1] = (int)((nelem & 0xFFFFu) << 16);      // tensor_dim0[15:0] @ bits 63:48
    g1[2] = (int)(((nelem >> 16) & 0xFFFFu)      // tensor_dim0[31:16]
                  | (1u << 16));                 // tensor_dim1 = 1
    g1[3] = (int)((nelem & 0xFFFFu) << 16);      // tile_dim0 = nelem (bits 127:112)
    g1[4] = 1;                                   // tile_dim1 = 1, tile_dim2 = 0
    g1[5] = (int)nelem;                          // tensor_dim0_stride[31:0]
    g1[6] = 0;                                   // stride hi / dim1_stride lo
    g1[7] = 0;

    v4i gz4 = {0, 0, 0, 0};                      // groups 2/3 unused (<=2-D)
    v8i gz8 = {0, 0, 0, 0, 0, 0, 0, 0};
    __builtin_amdgcn_tensor_load_to_lds(g0, g1, gz4, gz4, gz8, /*cpol=*/0);
}

// --------------------------------------------------------------------------
// Per-cell loss (everything except the box1 non-responsible term, which needs
// the next conobj cell's r and is handled by pass 2 via the flag array).
// flag: 0 = noobj cell, 1 = conobj r==0, 2 = conobj r==1.
// --------------------------------------------------------------------------
__device__ __forceinline__ float cell_loss(const float* __restrict__ p,
                                           const float* __restrict__ t,
                                           unsigned char* __restrict__ flagOut)
{
    const float t4 = t[4];
    if (t4 > 0.0f) {
        // class loss (channels 10..25)
        float cls = 0.0f;
#pragma unroll
        for (int k = 10; k < NVEC; ++k) {
            float d = p[k] - t[k];
            cls = fmaf(d, d, cls);
        }
        // per-box IoU (pred box b vs gt box b)
        float iou0 = 0.0f, iou1 = 0.0f;
#pragma unroll
        for (int b = 0; b < 2; ++b) {
            const float* pb = p + b * 5;
            const float* gb = t + b * 5;
            float pcx = pb[0] * GRID_LEN_F, pcy = pb[1] * GRID_LEN_F;
            float phx = HALF_IMG_F * pb[2], phy = HALF_IMG_F * pb[3];
            float gcx = gb[0] * GRID_LEN_F, gcy = gb[1] * GRID_LEN_F;
            float ghx = HALF_IMG_F * gb[2], ghy = HALF_IMG_F * gb[3];
            float pmnx = pcx - phx, pmxx = pcx + phx;
            float pmny = pcy - phy, pmxy = pcy + phy;
            float gmnx = gcx - ghx, gmxx = gcx + ghx;
            float gmny = gcy - ghy, gmxy = gcy + ghy;
            float ix = fmaxf(fminf(pmxx, gmxx) - fmaxf(pmnx, gmnx), 0.0f);
            float iy = fmaxf(fminf(pmxy, gmxy) - fmaxf(pmny, gmny), 0.0f);
            float ia = ix * iy;
            float pa = fmaxf(pmxx - pmnx, 0.0f) * fmaxf(pmxy - pmny, 0.0f);
            float ga = fmaxf(gmxx - gmnx, 0.0f) * fmaxf(gmxy - gmny, 0.0f);
            float v  = ia / (pa + ga - ia + 1e-6f);
            if (b == 0) iou0 = v; else iou1 = v;
        }
        // argmax: first index on ties -> r==1 only on strict >
        const int   r    = (iou1 > iou0) ? 1 : 0;
        const float miou = fmaxf(iou0, iou1);
        const float* pr = p + r * 5;
        const float* tr = t + r * 5;
        float dx = pr[0] - tr[0];
        float dy = pr[1] - tr[1];
        float dw = __builtin_sqrtf(pr[2]) - __builtin_sqrtf(tr[2]);
        float dh = __builtin_sqrtf(pr[3]) - __builtin_sqrtf(tr[3]);
        float loc  = dx * dx + dy * dy + dw * dw + dh * dh;
        float dc   = pr[4] - miou;
        float nr0  = r ? (p[4] * p[4]) : 0.0f;   // box0 non-responsible conf^2
        *flagOut = (unsigned char)(1 + r);
        return L_COORD_F * loc + dc * dc + nr0 + cls;
    }
    *flagOut = 0;
    if (t4 == 0.0f) {
        float d = p[4] - t4;
        return L_NOOBJ_F * d * d;
    }
    return 0.0f;
}

// --------------------------------------------------------------------------
// Pass 1: TDM double-buffered streaming over all cells.
// --------------------------------------------------------------------------
__global__ void __launch_bounds__(P1_BLOCK)
yolo_pass1(const float* __restrict__ pred, const float* __restrict__ targ,
           float* __restrict__ acc, unsigned char* __restrict__ flags,
           int ncell, int nchunks)
{
    __shared__ float ldsP[2][CHUNK * NVEC];
    __shared__ float ldsT[2][CHUNK * NVEC];
    __shared__ float red[P1_BLOCK];

    const int  tid   = (int)threadIdx.x;
    const bool wave0 = (tid < 32);           // wave-uniform predicate

    float sum = 0.0f;
    int   c   = (int)blockIdx.x;

    // Prologue: DMA first chunk into stage 0.
    if (c < nchunks && wave0) {
        int cells = ncell - c * CHUNK; if (cells > CHUNK) cells = CHUNK;
        uint32_t nel = (uint32_t)cells * NVEC;
        tdm_load_1d((uint32_t)(uintptr_t)&ldsP[0][0],
                    pred + (size_t)c * CHUNK * NVEC, nel);
        tdm_load_1d((uint32_t)(uintptr_t)&ldsT[0][0],
                    targ + (size_t)c * CHUNK * NVEC, nel);
        __builtin_amdgcn_s_wait_tensorcnt(0);
    }
    __syncthreads();

    int s = 0;
    for (; c < nchunks; c += (int)gridDim.x, s ^= 1) {
        // Issue next chunk's DMA into the other stage (overlaps with compute).
        int cn = c + (int)gridDim.x;
        if (cn < nchunks && wave0) {
            int cells2 = ncell - cn * CHUNK; if (cells2 > CHUNK) cells2 = CHUNK;
            uint32_t nel2 = (uint32_t)cells2 * NVEC;
            tdm_load_1d((uint32_t)(uintptr_t)&ldsP[s ^ 1][0],
                        pred + (size_t)cn * CHUNK * NVEC, nel2);
            tdm_load_1d((uint32_t)(uintptr_t)&ldsT[s ^ 1][0],
                        targ + (size_t)cn * CHUNK * NVEC, nel2);
        }
        // Consume current stage: one cell per thread.
        int cells = ncell - c * CHUNK; if (cells > CHUNK) cells = CHUNK;
        if (tid < cells) {
            unsigned char flag;
            float part = cell_loss(&ldsP[s][tid * NVEC], &ldsT[s][tid * NVEC], &flag);
            flags[(size_t)c * CHUNK + tid] = flag;
            sum += part;
        }
        if (wave0) __builtin_amdgcn_s_wait_tensorcnt(0);
        __syncthreads();
    }

    // Block reduction -> one fp32 global atomic per block.
    red[tid] = sum;
    __syncthreads();
    for (int off = P1_BLOCK / 2; off > 0; off >>= 1) {
        if (tid < off) red[tid] += red[tid + off];
        __syncthreads();
    }
    if (tid == 0) atomicAdd(acc, red[0]);
}

// --------------------------------------------------------------------------
// Pass 2: box1 non-responsible term. Cell i (conobj) contributes p[i,9]^2 iff
// the next conobj cell in flattened order (cyclic wrap) has r==0 (flag==1).
// Flags array (~0.8 MB) is L2-resident; expected scan length ~1/p_obj.
// --------------------------------------------------------------------------
__global__ void __launch_bounds__(P2_BLOCK)
yolo_pass2(const float* __restrict__ pred,
           const unsigned char* __restrict__ flags,
           float* __restrict__ acc, int ncell)
{
    __shared__ float red[P2_BLOCK];
    float sum = 0.0f;
    for (int i = (int)(blockIdx.x * P2_BLOCK + threadIdx.x); i < ncell;
         i += (int)(gridDim.x * P2_BLOCK)) {
        unsigned f = flags[i];
        if (f) {
            int k = i + 1; if (k == ncell) k = 0;
            unsigned fn = flags[k];
            while (!fn) {                       // terminates: flags[i] != 0
                ++k; if (k == ncell) k = 0;
                fn = flags[k];
            }
            if (fn == 1u) {                     // next conobj cell has r==0
                float c1 = pred[(size_t)i * NVEC + 9];
                sum = fmaf(c1, c1, sum);
            }
        }
    }
    red[threadIdx.x] = sum;
    __syncthreads();
    for (int off = P2_BLOCK / 2; off > 0; off >>= 1) {
        if ((int)threadIdx.x < off) red[threadIdx.x] += red[threadIdx.x + off];
        __syncthreads();
    }
    if (threadIdx.x == 0) atomicAdd(acc, red[0]);
}

__global__ void yolo_init(float* __restrict__ acc)
{
    if (threadIdx.x == 0 && blockIdx.x == 0) *acc = 0.0f;
}

__global__ void yolo_final(const float* __restrict__ acc, float* __restrict__ out,
                           float inv_nbatch)
{
    if (threadIdx.x == 0 && blockIdx.x == 0) *out = *acc * inv_nbatch;
}

// --------------------------------------------------------------------------
extern "C" void kernel_launch(void* const* d_in, const int* in_sizes, int n_in,
                              void* d_out, int out_size, void* d_ws, size_t ws_size,
                              hipStream_t stream)
{
    const float* pred = (const float*)d_in[0];
    const float* targ = (const float*)d_in[1];
    float* out = (float*)d_out;

    const int n      = in_sizes[0];        // N*7*7*26
    const int ncell  = n / NVEC;           // N*49
    const int nbatch = ncell / 49;         // N

    float*         acc   = (float*)d_ws;
    unsigned char* flags = (unsigned char*)d_ws + 256;

    const int nchunks = (ncell + CHUNK - 1) / CHUNK;
    int nb1 = nchunks < 2048 ? nchunks : 2048;
    int nb2 = (ncell + P2_BLOCK - 1) / P2_BLOCK;
    if (nb2 > 1024) nb2 = 1024;

    yolo_init <<<1, 1, 0, stream>>>(acc);
    yolo_pass1<<<nb1, P1_BLOCK, 0, stream>>>(pred, targ, acc, flags, ncell, nchunks);
    yolo_pass2<<<nb2, P2_BLOCK, 0, stream>>>(pred, flags, acc, ncell);
    yolo_final<<<1, 1, 0, stream>>>(acc, out, 1.0f / (float)nbatch);
}